// conv_approximation_51780125720678
// MI455X (gfx1250) — compile-verified
//
#include <hip/hip_runtime.h>
#include <math.h>

typedef float v2f __attribute__((ext_vector_type(2)));
typedef float v8f __attribute__((ext_vector_type(8)));

#define BB 16
#define NN 8192
#define HH 4
#define DD 32
#define MM 128
#define NCHUNK 128
#define CHUNKS (NN / NCHUNK)   // 64

#define NORMALIZER 0.42044820762685725f  // 32^(-1/4)
#define RATIO      0.08838834764831845f  // 1/sqrt(128)
#define SCALE      NORMALIZER            // * inv_sqrt_tau (tau = 1)
#define FEPS       1e-4f

// ---- order-preserving float<->uint encoding for atomic max ----
__device__ __forceinline__ unsigned encodeF(float f) {
    unsigned u = __float_as_uint(f);
    return (u & 0x80000000u) ? ~u : (u | 0x80000000u);
}
__device__ __forceinline__ float decodeF(unsigned e) {
    unsigned u = (e & 0x80000000u) ? (e & 0x7FFFFFFFu) : ~e;
    return __uint_as_float(u);
}

// ---- projection GEMM: dd[128 rows][128 m] = sU @ sProj^T via V_WMMA_F32_16X16X4_F32
// wave w owns row tile [w*16, w*16+16), all 8 m-tiles. K = 32 in steps of 4.
__device__ __forceinline__ void proj_gemm(const float (*sU)[DD],
                                          const float (*sProj)[DD],
                                          v8f acc[8], int lane, int w) {
    const int half = lane >> 4;
    const int lm   = lane & 15;
    const int r    = w * 16 + lm;
#pragma unroll
    for (int kb = 0; kb < 8; ++kb) {
        const int k = kb * 4 + half * 2;
        v2f a;  // A 16x4 f32: lanes0-15 hold K=k..k+1 of row M=lm; lanes16-31 K=k+2..k+3
        a.x = sU[r][k];
        a.y = sU[r][k + 1];
#pragma unroll
        for (int mt = 0; mt < 8; ++mt) {
            v2f b;  // B 4x16: row K striped across lanes (N = m)
            b.x = sProj[mt * 16 + lm][k];
            b.y = sProj[mt * 16 + lm][k + 1];
            acc[mt] = __builtin_amdgcn_wmma_f32_16x16x4_f32(
                false, a, false, b, (short)0, acc[mt], false, false);
        }
    }
}

// ---- coalesced float4 store of a 128x128 LDS tile to [.., n, h, m] layout ----
__device__ __forceinline__ void store_rows(float* __restrict__ dstBase,
                                           const float (*sP)[MM + 2], int t) {
#pragma unroll
    for (int i = 0; i < 16; ++i) {
        const int f = t + 256 * i;      // float4 index, 4096 total
        const int n = f >> 5;
        const int c = (f & 31) << 2;
        float4 v = make_float4(sP[n][c], sP[n][c + 1], sP[n][c + 2], sP[n][c + 3]);
        *(float4*)(dstBase + (size_t)n * (HH * MM) + c) = v;
    }
}

__global__ void k_init(unsigned* ws, int n) {
    int i = blockIdx.x * 256 + threadIdx.x;
    if (i < n) ws[i] = 0u;
}

// ================= pass 1: global max of data_dash2 (key stabilizer) =================
__global__ __launch_bounds__(256) void k_maxkey(const float* __restrict__ nv2,
                                                const float* __restrict__ proj,
                                                unsigned* __restrict__ wsMax) {
    __shared__ __align__(16) float sProj[MM][DD];
    __shared__ __align__(16) float sU[NCHUNK][DD];
    __shared__ float sRed[256];

    const int t = threadIdx.x, lane = t & 31, w = t >> 5;
    const int nc = blockIdx.x & (CHUNKS - 1);
    const int bh = blockIdx.x >> 6;
    const int b = bh >> 2, h = bh & 3;
    const int nbase = nc * NCHUNK;

    {   const float4* s = (const float4*)proj;
        float4* d = (float4*)&sProj[0][0];
#pragma unroll
        for (int i = 0; i < 4; ++i) d[t + 256 * i] = s[t + 256 * i];
    }
    if (t < NCHUNK) {
        const float4* s = (const float4*)(nv2 + ((size_t)((size_t)b * NN + nbase + t) * HH + h) * DD);
        float4* d = (float4*)&sU[t][0];
#pragma unroll
        for (int i = 0; i < 8; ++i) {
            float4 v = s[i];
            v.x *= SCALE; v.y *= SCALE; v.z *= SCALE; v.w *= SCALE;
            d[i] = v;
        }
    }
    __syncthreads();

    const v8f vzero = {0.f, 0.f, 0.f, 0.f, 0.f, 0.f, 0.f, 0.f};
    v8f acc[8];
#pragma unroll
    for (int mt = 0; mt < 8; ++mt) acc[mt] = vzero;
    proj_gemm(sU, sProj, acc, lane, w);

    float m = -3.4e38f;
#pragma unroll
    for (int mt = 0; mt < 8; ++mt)
#pragma unroll
        for (int j = 0; j < 8; ++j) m = fmaxf(m, acc[mt][j]);
    sRed[t] = m;
    __syncthreads();
    for (int s = 128; s > 0; s >>= 1) {
        if (t < s) sRed[t] = fmaxf(sRed[t], sRed[t + s]);
        __syncthreads();
    }
    if (t == 0) atomicMax(wsMax, encodeF(sRed[0]));
}

// ================= pass 2: p2, v2x += p2^T @ x, z += colsum(p2) =================
__global__ __launch_bounds__(256) void k_key(const float* __restrict__ nv2,
                                             const float* __restrict__ x,
                                             const float* __restrict__ proj,
                                             const unsigned* __restrict__ wsMax,
                                             float* __restrict__ p2out,
                                             float* __restrict__ wsV2x,
                                             float* __restrict__ wsZ) {
    __shared__ __align__(16) float sProj[MM][DD];
    __shared__ __align__(16) float sU[NCHUNK][DD];
    __shared__ __align__(16) float sX[NCHUNK][DD];
    __shared__ float sDiag[NCHUNK];
    __shared__ float sP[NCHUNK][MM + 2];   // padded: conflict-free column walks

    const int t = threadIdx.x, lane = t & 31, w = t >> 5;
    const int nc = blockIdx.x & (CHUNKS - 1);
    const int bh = blockIdx.x >> 6;
    const int b = bh >> 2, h = bh & 3;
    const int nbase = nc * NCHUNK;

    {   const float4* s = (const float4*)proj;
        float4* d = (float4*)&sProj[0][0];
#pragma unroll
        for (int i = 0; i < 4; ++i) d[t + 256 * i] = s[t + 256 * i];
    }
    if (t < NCHUNK) {
        const float4* s = (const float4*)(nv2 + ((size_t)((size_t)b * NN + nbase + t) * HH + h) * DD);
        float4* d = (float4*)&sU[t][0];
        float ss = 0.f;
#pragma unroll
        for (int i = 0; i < 8; ++i) {
            float4 v = s[i];
            v.x *= SCALE; v.y *= SCALE; v.z *= SCALE; v.w *= SCALE;
            ss += v.x * v.x + v.y * v.y + v.z * v.z + v.w * v.w;
            d[i] = v;
        }
        sDiag[t] = 0.5f * ss;   // diag = 0.5*norm^2*sum(data^2) = 0.5*sum(u^2)
    } else {
        const int r = t - NCHUNK;
        const float4* s = (const float4*)(x + ((size_t)((size_t)b * NN + nbase + r) * HH + h) * DD);
        float4* d = (float4*)&sX[r][0];
#pragma unroll
        for (int i = 0; i < 8; ++i) d[i] = s[i];
    }
    __syncthreads();

    const v8f vzero = {0.f, 0.f, 0.f, 0.f, 0.f, 0.f, 0.f, 0.f};
    v8f acc[8];
#pragma unroll
    for (int mt = 0; mt < 8; ++mt) acc[mt] = vzero;
    proj_gemm(sU, sProj, acc, lane, w);

    const float stab = decodeF(*wsMax);
    const int half = lane >> 4, lm = lane & 15;
#pragma unroll
    for (int mt = 0; mt < 8; ++mt)
#pragma unroll
        for (int j = 0; j < 8; ++j) {
            const int r = w * 16 + j + 8 * half;   // D-matrix VGPR j -> rows j / j+8
            const float p = RATIO * (__expf(acc[mt][j] - sDiag[r] - stab) + FEPS);
            sP[r][mt * 16 + lm] = p;
        }
    __syncthreads();

    store_rows(p2out + ((size_t)((size_t)b * NN + nbase) * HH + h) * MM, sP, t);

    // v2x partial: [m=128][d=32] = p2tile^T @ xtile, K = 128 rows. wave w = m-tile.
    v8f a0 = vzero, a1 = vzero;
#pragma unroll
    for (int kb = 0; kb < 32; ++kb) {
        const int k = kb * 4 + half * 2;           // k indexes n within the chunk
        v2f av; av.x = sP[k][w * 16 + lm]; av.y = sP[k + 1][w * 16 + lm];
        v2f b0; b0.x = sX[k][lm];          b0.y = sX[k + 1][lm];
        v2f b1; b1.x = sX[k][16 + lm];     b1.y = sX[k + 1][16 + lm];
        a0 = __builtin_amdgcn_wmma_f32_16x16x4_f32(false, av, false, b0, (short)0, a0, false, false);
        a1 = __builtin_amdgcn_wmma_f32_16x16x4_f32(false, av, false, b1, (short)0, a1, false, false);
    }
    float* vbase = wsV2x + (size_t)bh * MM * DD;
#pragma unroll
    for (int j = 0; j < 8; ++j) {
        const int m = w * 16 + j + 8 * half;
        atomicAdd(&vbase[m * DD + lm],      a0[j]);
        atomicAdd(&vbase[m * DD + 16 + lm], a1[j]);
    }
    // z partial
    {   const int m = t & 127, nh = t >> 7;
        float s = 0.f;
#pragma unroll 8
        for (int i = 0; i < 64; ++i) s += sP[nh * 64 + i][m];
        atomicAdd(&wsZ[bh * MM + m], s);
    }
}

// ================= pass 3: p1, out = (p1 @ v2x) / (p1 @ z) =================
__global__ __launch_bounds__(256) void k_query(const float* __restrict__ nv1,
                                               const float* __restrict__ proj,
                                               const float* __restrict__ wsV2x,
                                               const float* __restrict__ wsZ,
                                               float* __restrict__ p1out,
                                               float* __restrict__ outp) {
    __shared__ __align__(16) float sProj[MM][DD];
    __shared__ __align__(16) float sU[NCHUNK][DD];
    __shared__ __align__(16) float sV[MM][DD];     // v2x slice for this (b,h)
    __shared__ float sDiag[NCHUNK];
    __shared__ float sZ[MM];
    __shared__ float sDen[NCHUNK];
    __shared__ float sP[NCHUNK][MM + 2];

    const int t = threadIdx.x, lane = t & 31, w = t >> 5;
    const int nc = blockIdx.x & (CHUNKS - 1);
    const int bh = blockIdx.x >> 6;
    const int b = bh >> 2, h = bh & 3;
    const int nbase = nc * NCHUNK;

    {   const float4* s = (const float4*)proj;
        float4* d = (float4*)&sProj[0][0];
#pragma unroll
        for (int i = 0; i < 4; ++i) d[t + 256 * i] = s[t + 256 * i];
    }
    if (t < NCHUNK) {
        const float4* s = (const float4*)(nv1 + ((size_t)((size_t)b * NN + nbase + t) * HH + h) * DD);
        float4* d = (float4*)&sU[t][0];
        float ss = 0.f;
#pragma unroll
        for (int i = 0; i < 8; ++i) {
            float4 v = s[i];
            v.x *= SCALE; v.y *= SCALE; v.z *= SCALE; v.w *= SCALE;
            ss += v.x * v.x + v.y * v.y + v.z * v.z + v.w * v.w;
            d[i] = v;
        }
        sDiag[t] = 0.5f * ss;
        sZ[t] = wsZ[bh * MM + t];
    } else {
        const int tt = t - NCHUNK;
        const float4* s = (const float4*)(wsV2x + (size_t)bh * MM * DD);
        float4* d = (float4*)&sV[0][0];
#pragma unroll
        for (int i = 0; i < 8; ++i) d[tt + 128 * i] = s[tt + 128 * i];
    }
    __syncthreads();

    const v8f vzero = {0.f, 0.f, 0.f, 0.f, 0.f, 0.f, 0.f, 0.f};
    v8f acc[8];
#pragma unroll
    for (int mt = 0; mt < 8; ++mt) acc[mt] = vzero;
    proj_gemm(sU, sProj, acc, lane, w);

    const int half = lane >> 4, lm = lane & 15;
#pragma unroll
    for (int mt = 0; mt < 8; ++mt)
#pragma unroll
        for (int j = 0; j < 8; ++j)
            sP[w * 16 + j + 8 * half][mt * 16 + lm] = acc[mt][j];
    __syncthreads();

    if (t < NCHUNK) {           // per-row stabilizer, p1, and denominator dot
        float rmax = -3.4e38f;
#pragma unroll 8
        for (int m = 0; m < MM; ++m) rmax = fmaxf(rmax, sP[t][m]);
        const float dg = sDiag[t];
        float den = 0.f;
#pragma unroll 4
        for (int m = 0; m < MM; ++m) {
            const float p = RATIO * (__expf(sP[t][m] - dg - rmax) + FEPS);
            sP[t][m] = p;
            den += p * sZ[m];
        }
        sDen[t] = den;
    }
    __syncthreads();

    store_rows(p1out + ((size_t)((size_t)b * NN + nbase) * HH + h) * MM, sP, t);

    // out1 = p1 @ v2x : [n=128][d=32], K = m = 128. wave w = n-tile.
    v8f a0 = vzero, a1 = vzero;
#pragma unroll
    for (int kb = 0; kb < 32; ++kb) {
        const int k = kb * 4 + half * 2;           // k indexes m
        v2f av; av.x = sP[w * 16 + lm][k]; av.y = sP[w * 16 + lm][k + 1];
        v2f b0; b0.x = sV[k][lm];          b0.y = sV[k + 1][lm];
        v2f b1; b1.x = sV[k][16 + lm];     b1.y = sV[k + 1][16 + lm];
        a0 = __builtin_amdgcn_wmma_f32_16x16x4_f32(false, av, false, b0, (short)0, a0, false, false);
        a1 = __builtin_amdgcn_wmma_f32_16x16x4_f32(false, av, false, b1, (short)0, a1, false, false);
    }
    float* obase = outp + ((size_t)((size_t)b * NN + nbase) * HH + h) * DD;
#pragma unroll
    for (int j = 0; j < 8; ++j) {
        const int r = w * 16 + j + 8 * half;
        const float inv = 1.0f / sDen[r];
        obase[(size_t)r * (HH * DD) + lm]      = a0[j] * inv;
        obase[(size_t)r * (HH * DD) + 16 + lm] = a1[j] * inv;
    }
}

extern "C" void kernel_launch(void* const* d_in, const int* in_sizes, int n_in,
                              void* d_out, int out_size, void* d_ws, size_t ws_size,
                              hipStream_t stream) {
    (void)in_sizes; (void)n_in; (void)out_size; (void)ws_size;
    const float* x    = (const float*)d_in[0];
    const float* nv1  = (const float*)d_in[1];
    const float* nv2  = (const float*)d_in[2];
    const float* proj = (const float*)d_in[3];

    float* outp = (float*)d_out;                               // [B,N,H,D]
    float* p1o  = outp + (size_t)BB * NN * HH * DD;            // [B,N,H,M]
    float* p2o  = p1o + (size_t)BB * NN * HH * MM;             // [B,N,H,M]

    unsigned* wsMax = (unsigned*)d_ws;                         // slot 0 (encoded max)
    float* wsV2x = (float*)d_ws + 64;                          // [B,H,M,D] = 262144
    float* wsZ   = wsV2x + (size_t)BB * HH * MM * DD;          // [B,H,M]   = 8192

    const int wsInit = 64 + BB * HH * MM * DD + BB * HH * MM;  // 270400 dwords
    k_init<<<(wsInit + 255) / 256, 256, 0, stream>>>((unsigned*)d_ws, wsInit);

    const dim3 grid(BB * HH * CHUNKS);                         // 4096 blocks
    k_maxkey<<<grid, 256, 0, stream>>>(nv2, proj, wsMax);
    k_key   <<<grid, 256, 0, stream>>>(nv2, x, proj, wsMax, p2o, wsV2x, wsZ);
    k_query <<<grid, 256, 0, stream>>>(nv1, proj, wsV2x, wsZ, p1o, outp);
}